// WavePool_8804682957394
// MI455X (gfx1250) — compile-verified
//
#include <hip/hip_runtime.h>

// Problem constants (fixed by the reference setup)
#define NNODES   131072     // N
#define FDIM     512        // F
#define NEDGES   2097152    // E
#define NP       65536      // pooled node count (N/2)
#define FP       256        // pooled feature dim (F/2)
#define SORT_MAX 2048       // per-bucket cap (avg occupancy is 32; Poisson tail << 2048)

typedef __attribute__((ext_vector_type(2))) float v2f;
typedef __attribute__((ext_vector_type(8))) float v8f;

// ---------------------------------------------------------------------------
// Kernel 1: 2x2 average pool of x (131072x512 -> 65536x256) via f32 WMMA.
// Each wave computes one 16x16 tile of x_pooled:
//   D = sum_j A_j (16x4) * B_j (4x16), A = row-pair-summed x, B = 0.25-selector.
// ---------------------------------------------------------------------------
__global__ void pool_x_wmma(const float* __restrict__ x, float* __restrict__ out) {
  const int wave = (blockIdx.x * blockDim.x + threadIdx.x) >> 5; // 0..65535
  const int lane = threadIdx.x & 31;
  const int rt = wave >> 4;       // row tile (0..4095), 16 pooled rows each
  const int ct = wave & 15;       // col tile (0..15), 16 pooled cols each
  const int m   = lane & 15;      // A-row / B-col held by this lane
  const int grp = lane >> 4;      // 0: K=0,1   1: K=2,3

  const long r0 = (long)rt * 16;                 // pooled row base
  const float* p0 = x + ((r0 + m) * 2) * FDIM;   // original row 2n
  const float* p1 = p0 + FDIM;                   // original row 2n+1
  const int c0 = ct * 32;                        // original col base for this tile

  v8f acc = {};
#pragma unroll
  for (int j = 0; j < 8; ++j) {
    const int cc = c0 + 4 * j + 2 * grp;         // this lane's K columns (even -> 8B aligned)
    v2f a0 = *(const v2f*)(p0 + cc);
    v2f a1 = *(const v2f*)(p1 + cc);
    v2f A  = a0 + a1;                            // row-pair sum
    // B[k,n] = 0.25 iff n == 2j + (k>>1); this lane holds k = 2*grp, 2*grp+1 (same k>>1)
    const float bv = (m == 2 * j + grp) ? 0.25f : 0.0f;
    v2f B = { bv, bv };
    acc = __builtin_amdgcn_wmma_f32_16x16x4_f32(false, A, false, B,
                                                (short)0, acc, false, false);
  }
  // D layout: VGPR v -> row r0 + v + 8*grp, col ct*16 + m
  float* orow = out + (r0 + grp * 8) * FP + ct * 16 + m;
#pragma unroll
  for (int v = 0; v < 8; ++v)
    orow[(long)v * FP] = acc[v];
}

// ---------------------------------------------------------------------------
// Kernel 2: zero the small counter arrays (must re-zero every launch).
// ---------------------------------------------------------------------------
__global__ void init_ws(unsigned* hist, unsigned* cursor, unsigned* ucount) {
  const int i = blockIdx.x * blockDim.x + threadIdx.x;
  if (i < NP) { hist[i] = 0u; cursor[i] = 0u; ucount[i] = 0u; }
}

// ---------------------------------------------------------------------------
// Kernel 3: default fill for pooled edge outputs + batch_pooled.
// Fill rows correspond to unique() sentinel = Np*Np: idx = (65536, 0), attr = 0.
// ---------------------------------------------------------------------------
__global__ void fill_defaults(float* oIdxR, float* oIdxC, float* oAttr, float* oBatch) {
  const int i = blockIdx.x * blockDim.x + threadIdx.x;
  if (i >= NEDGES) return;
  oIdxR[i] = 65536.0f;
  oIdxC[i] = 0.0f;
  float4 z = {0.f, 0.f, 0.f, 0.f};
  float4* a = (float4*)(oAttr + (size_t)i * 16);
  a[0] = z; a[1] = z; a[2] = z; a[3] = z;
  if (i < NP) oBatch[i] = (float)(i >> 10);   // batch[::2] = i // 1024
}

// ---------------------------------------------------------------------------
// Kernel 4: histogram of row_p (top 16 bits of the 32-bit pooled key).
// ---------------------------------------------------------------------------
__global__ void edge_hist(const long long* __restrict__ ei, unsigned* hist) {
  const int e = blockIdx.x * blockDim.x + threadIdx.x;
  if (e >= NEDGES) return;
  const unsigned r = ((unsigned)ei[e]) >> 1;
  atomicAdd(&hist[r], 1u);
}

// ---------------------------------------------------------------------------
// Kernel 5: exclusive scan of 65536 uints, single 1024-thread block,
// chunked Hillis-Steele. out has n+1 entries (total at out[n]).
// ---------------------------------------------------------------------------
__global__ void scan_kernel(const unsigned* __restrict__ in, unsigned* out, int n) {
  __shared__ unsigned s[1024];
  __shared__ unsigned carry;
  const int tid = threadIdx.x;
  if (tid == 0) carry = 0u;
  __syncthreads();
  for (int base = 0; base < n; base += 1024) {
    const unsigned v = in[base + tid];
    s[tid] = v;
    __syncthreads();
    for (int off = 1; off < 1024; off <<= 1) {
      const unsigned t = (tid >= off) ? s[tid - off] : 0u;
      __syncthreads();
      s[tid] += t;
      __syncthreads();
    }
    out[base + tid] = carry + s[tid] - v;   // exclusive
    __syncthreads();
    if (tid == 0) carry += s[1023];
    __syncthreads();
  }
  if (tid == 0) out[n] = carry;
}

// ---------------------------------------------------------------------------
// Kernel 6: scatter edges into row_p buckets as packed (col_p<<32 | edge_id).
// ---------------------------------------------------------------------------
__global__ void edge_scatter(const long long* __restrict__ ei,
                             const unsigned* __restrict__ start,
                             unsigned* cursor, unsigned long long* keybuf) {
  const int e = blockIdx.x * blockDim.x + threadIdx.x;
  if (e >= NEDGES) return;
  const unsigned r = ((unsigned)ei[e]) >> 1;
  const unsigned c = ((unsigned)ei[NEDGES + e]) >> 1;
  const unsigned pos = atomicAdd(&cursor[r], 1u);
  keybuf[start[r] + pos] = ((unsigned long long)c << 32) | (unsigned)e;
}

// ---------------------------------------------------------------------------
// Kernel 7: per-bucket bitonic sort by (col_p, edge_id) in LDS + unique count.
// One 128-thread block per bucket; canonical order => deterministic sums.
// ---------------------------------------------------------------------------
__global__ void bucket_sort(unsigned long long* keybuf,
                            const unsigned* __restrict__ hist,
                            const unsigned* __restrict__ start,
                            unsigned* ucount) {
  __shared__ unsigned long long s[SORT_MAX];
  __shared__ unsigned ured;
  const int b   = blockIdx.x;
  const int tid = threadIdx.x;
  unsigned cnt = hist[b];
  if (cnt > SORT_MAX) cnt = SORT_MAX;
  if (cnt == 0) { if (tid == 0) ucount[b] = 0u; return; }
  const unsigned s0 = start[b];
  unsigned P = 1; while (P < cnt) P <<= 1;

  for (unsigned i = tid; i < P; i += blockDim.x)
    s[i] = (i < cnt) ? keybuf[s0 + i] : 0xFFFFFFFFFFFFFFFFULL;
  __syncthreads();

  for (unsigned k = 2; k <= P; k <<= 1) {
    for (unsigned j = k >> 1; j > 0; j >>= 1) {
      for (unsigned i = tid; i < P; i += blockDim.x) {
        const unsigned ixj = i ^ j;
        if (ixj > i) {
          const bool asc = ((i & k) == 0);
          const unsigned long long a = s[i], c = s[ixj];
          if (asc ? (a > c) : (a < c)) { s[i] = c; s[ixj] = a; }
        }
      }
      __syncthreads();
    }
  }

  if (tid == 0) ured = 0u;
  __syncthreads();
  unsigned lu = 0;
  for (unsigned i = tid; i < cnt; i += blockDim.x) {
    const unsigned long long k = s[i];
    keybuf[s0 + i] = k;                       // write back sorted
    const unsigned c = (unsigned)(k >> 32);
    if (i == 0 || c != (unsigned)(s[i - 1] >> 32)) lu++;
  }
  atomicAdd(&ured, lu);
  __syncthreads();
  if (tid == 0) ucount[b] = ured;
}

// ---------------------------------------------------------------------------
// Kernel 8: one thread per bucket walks its sorted slice, emits unique keys
// and serially-summed attrs (reference summation order: increasing edge id).
// ---------------------------------------------------------------------------
__global__ void emit_kernel(const unsigned long long* __restrict__ keybuf,
                            const unsigned* __restrict__ hist,
                            const unsigned* __restrict__ start,
                            const unsigned* __restrict__ ubase,
                            const float* __restrict__ attr,
                            float* oIdxR, float* oIdxC, float* oAttr) {
  const int b = blockIdx.x * blockDim.x + threadIdx.x;
  if (b >= NP) return;
  unsigned cnt = hist[b];
  if (cnt > SORT_MAX) cnt = SORT_MAX;
  const unsigned s0 = start[b];
  unsigned orow = ubase[b];
  float acc[16];
  unsigned prevc = 0u;
  bool have = false;
  for (unsigned i = 0; i < cnt; ++i) {
    const unsigned long long k = keybuf[s0 + i];
    const unsigned c = (unsigned)(k >> 32);
    const unsigned e = (unsigned)k;
    const float* ar = attr + (size_t)e * 16;
    if (!have || c != prevc) {
      if (have) {
        oIdxR[orow] = (float)b;
        oIdxC[orow] = (float)prevc;
        float* dst = oAttr + (size_t)orow * 16;
        for (int q = 0; q < 16; ++q) dst[q] = acc[q];
        orow++;
      }
      for (int q = 0; q < 16; ++q) acc[q] = ar[q];
      prevc = c; have = true;
    } else {
      for (int q = 0; q < 16; ++q) acc[q] += ar[q];
    }
  }
  if (have) {
    oIdxR[orow] = (float)b;
    oIdxC[orow] = (float)prevc;
    float* dst = oAttr + (size_t)orow * 16;
    for (int q = 0; q < 16; ++q) dst[q] = acc[q];
  }
}

// ---------------------------------------------------------------------------
extern "C" void kernel_launch(void* const* d_in, const int* in_sizes, int n_in,
                              void* d_out, int out_size, void* d_ws, size_t ws_size,
                              hipStream_t stream) {
  (void)in_sizes; (void)n_in; (void)out_size; (void)ws_size;
  const float*     x     = (const float*)d_in[0];       // (131072, 512) f32
  const float*     eattr = (const float*)d_in[1];       // (2097152, 16) f32
  const long long* eidx  = (const long long*)d_in[2];   // (2, 2097152) i64

  // Output layout (flat, return order): x_pooled | edge_index_pooled | edge_attr_pooled | batch_pooled
  float* out    = (float*)d_out;
  float* oX     = out;                                  // 65536*256
  float* oIdxR  = out + (size_t)NP * FP;                // E
  float* oIdxC  = oIdxR + NEDGES;                       // E
  float* oAttr  = oIdxC + NEDGES;                       // E*16
  float* oBatch = oAttr + (size_t)NEDGES * 16;          // NP

  // Workspace layout (~18 MB)
  char* ws = (char*)d_ws;
  unsigned long long* keybuf = (unsigned long long*)ws;            // E * 8B
  unsigned* hist   = (unsigned*)(ws + (size_t)NEDGES * 8);         // NP
  unsigned* start  = hist + NP;                                    // NP+1
  unsigned* cursor = start + NP + 1;                               // NP
  unsigned* ucount = cursor + NP;                                  // NP
  unsigned* ubase  = ucount + NP;                                  // NP+1

  init_ws      <<<NP / 256,      256, 0, stream>>>(hist, cursor, ucount);
  fill_defaults<<<NEDGES / 256,  256, 0, stream>>>(oIdxR, oIdxC, oAttr, oBatch);
  pool_x_wmma  <<<(NP / 16) * (FP / 16) * 32 / 256, 256, 0, stream>>>(x, oX); // 65536 waves
  edge_hist    <<<NEDGES / 256,  256, 0, stream>>>(eidx, hist);
  scan_kernel  <<<1,            1024, 0, stream>>>(hist, start, NP);
  edge_scatter <<<NEDGES / 256,  256, 0, stream>>>(eidx, start, cursor, keybuf);
  bucket_sort  <<<NP,            128, 0, stream>>>(keybuf, hist, start, ucount);
  scan_kernel  <<<1,            1024, 0, stream>>>(ucount, ubase, NP);
  emit_kernel  <<<NP / 256,      256, 0, stream>>>(keybuf, hist, start, ubase,
                                                   eattr, oIdxR, oIdxC, oAttr);
}